// NaturalCubicSplinePotential_57166014710223
// MI455X (gfx1250) — compile-verified
//
#include <hip/hip_runtime.h>
#include <hip/hip_bf16.h>

typedef __attribute__((ext_vector_type(2))) float v2f;
typedef __attribute__((ext_vector_type(8))) float v8f;

#define NM 8          // marginals
#define NN 128        // nodes
#define NI 127        // intervals
#define CO_STRIDE 128 // padded per-marginal coefficient stride (float4 entries)
#define EVAL_BLOCKS 4096
#define EVAL_THREADS 256

// ---------------------------------------------------------------------------
// Kernel 1: natural cubic spline coefficients via Thomas algorithm.
// System: c[0]=c[127]=0 ; c[i-1] + 4 c[i] + c[i+1] = 3(v[i-1]-2v[i]+v[i+1])/h^2
// One lane per marginal (8 active lanes). Negligible work.
// ---------------------------------------------------------------------------
__global__ void spline_coeff_kernel(const float* __restrict__ nodal,
                                    float4* __restrict__ coef) {
    __shared__ float sc[NM][NN];
    __shared__ float scp[NM][NN];
    __shared__ float sdp[NM][NN];
    const int f = threadIdx.x;
    if (f < NM) {
        const float h = 8.0f / 127.0f;
        const float inv_h2 = (127.0f * 127.0f) / 64.0f;
        const float* v = nodal + f * NN;

        // forward sweep
        float beta = 4.0f;
        float r = 3.0f * (v[0] - 2.0f * v[1] + v[2]) * inv_h2;
        scp[f][1] = 1.0f / beta;
        sdp[f][1] = r / beta;
        for (int i = 2; i <= 126; ++i) {
            r = 3.0f * (v[i - 1] - 2.0f * v[i] + v[i + 1]) * inv_h2;
            beta = 4.0f - scp[f][i - 1];
            scp[f][i] = 1.0f / beta;
            sdp[f][i] = (r - sdp[f][i - 1]) / beta;
        }
        // back substitution
        sc[f][0] = 0.0f;
        sc[f][127] = 0.0f;
        sc[f][126] = sdp[f][126];
        for (int i = 125; i >= 1; --i)
            sc[f][i] = sdp[f][i] - scp[f][i] * sc[f][i + 1];

        // per-interval coefficients a,b,c,d
        for (int i = 0; i < NI; ++i) {
            float ci  = sc[f][i];
            float ci1 = sc[f][i + 1];
            float b = (v[i + 1] - v[i]) / h - h * (2.0f * ci + ci1) * (1.0f / 3.0f);
            float d = (ci1 - ci) / (3.0f * h);
            coef[f * CO_STRIDE + i] = make_float4(v[i], b, ci, d);
        }
    }
}

// ---------------------------------------------------------------------------
// Kernel 2: streaming evaluation + reduction to per-block partials.
// ---------------------------------------------------------------------------
__device__ __forceinline__ float eval_spline(float xx, const float4* __restrict__ cf) {
    const float h = 8.0f / 127.0f;
    float xs = (xx + 4.0f) * (127.0f / 8.0f);     // x_scaled * (n-1)
    int idx = (int)ceilf(xs) - 1;
    idx = min(max(idx, 0), 126);
    float y = xx - (-4.0f + h * (float)idx);      // x - nodes[idx]
    float4 q = cf[idx];                            // ds_load_b128 from LDS
    return q.x + y * (q.y + y * (q.z + y * q.w)); // Horner
}

// ds_swizzle_b32, group-of-32 XOR mode: offset = (xor_mask<<10) | and_mask 0x1F
__device__ __forceinline__ float swz_add(float s, const int imm) {
    // caller passes a literal; wrapper exists only for readability
    return s;
}

__global__ void __launch_bounds__(EVAL_THREADS)
spline_eval_kernel(const float* __restrict__ x,
                   const float4* __restrict__ coef,
                   float* __restrict__ partials, int n4) {
    __shared__ float4 lcoef[NM * CO_STRIDE];   // 16 KB of 320 KB LDS
    __shared__ float wavesum[EVAL_THREADS / 32];

    // Cache all spline coefficients in LDS (random-index gathers hit LDS only).
    for (int i = threadIdx.x; i < NM * CO_STRIDE; i += blockDim.x)
        lcoef[i] = coef[i];
    __syncthreads();

    const float4* __restrict__ x4 = (const float4*)x;
    const int tid = blockIdx.x * blockDim.x + threadIdx.x;
    const int stride = gridDim.x * blockDim.x;

    // Two accumulators per lane -> exactly the 4x16 f32 B-matrix of one wave.
    v2f acc = {0.0f, 0.0f};

    // Uniform trip count: no exec-mask manipulation in the hot loop.
    const int iters = n4 / stride;               // uniform (SALU)
    for (int k = 0; k < iters; ++k) {
        const int i = tid + k * stride;
        if (k + 1 < iters)                       // uniform branch (SALU only)
            __builtin_prefetch(&x4[i + stride], 0, 3);  // global_prefetch, WGP scope
        float4 xv = x4[i];
        // element index e = 4*i ; marginal f = (e>>16)&7 = (i>>14)&7
        const float4* cf = &lcoef[((i >> 14) & 7) << 7];
        acc.x += eval_spline(xv.x, cf);
        acc.y += eval_spline(xv.y, cf);
        acc.x += eval_spline(xv.z, cf);
        acc.y += eval_spline(xv.w, cf);
    }
    // Tail (empty for the harness sizes; divergence ends before the WMMA).
    {
        const int i = iters * stride + tid;
        if (i < n4) {
            float4 xv = x4[i];
            const float4* cf = &lcoef[((i >> 14) & 7) << 7];
            acc.x += eval_spline(xv.x, cf);
            acc.y += eval_spline(xv.y, cf);
            acc.x += eval_spline(xv.z, cf);
            acc.y += eval_spline(xv.w, cf);
        }
    }

    // --- wave reduction via V_WMMA_F32_16X16X4_F32 ---------------------------
    // B (4x16 f32, 2 VGPRs) = per-lane partials; A = ones (16x4).
    // With A all-ones, D[0][n] = sum_k B[k][n]; summing D[0][0..15] equals the
    // sum of ALL 64 B elements regardless of the exact lane->slot striping.
    // Uniform control flow here => EXEC all ones (WMMA requirement).
    v2f ones = {1.0f, 1.0f};
    v8f czero = {};
    v8f dmat = __builtin_amdgcn_wmma_f32_16x16x4_f32(
        /*neg_a=*/false, ones, /*neg_b=*/false, acc,
        /*c_mod=*/(short)0, czero, /*reuse_a=*/false, /*reuse_b=*/false);

    // Sum the 16 column sums with ds_swizzle XOR exchanges (single DS op each).
    float s = dmat[0];
    s += __int_as_float(__builtin_amdgcn_ds_swizzle(__float_as_int(s), 0x041F)); // xor 1
    s += __int_as_float(__builtin_amdgcn_ds_swizzle(__float_as_int(s), 0x081F)); // xor 2
    s += __int_as_float(__builtin_amdgcn_ds_swizzle(__float_as_int(s), 0x101F)); // xor 4
    s += __int_as_float(__builtin_amdgcn_ds_swizzle(__float_as_int(s), 0x201F)); // xor 8

    const int lane = threadIdx.x & 31;
    const int wave = threadIdx.x >> 5;
    if (lane == 0) wavesum[wave] = s;
    __syncthreads();

    if (threadIdx.x == 0) {
        float bs = 0.0f;
        for (int w = 0; w < EVAL_THREADS / 32; ++w) bs += wavesum[w]; // fixed order
        partials[blockIdx.x] = bs;
    }
}

// ---------------------------------------------------------------------------
// Kernel 3: deterministic final reduction of per-block partials.
// ---------------------------------------------------------------------------
__global__ void spline_reduce_kernel(const float* __restrict__ partials, int n,
                                     float* __restrict__ out) {
    __shared__ float sdata[256];
    float s = 0.0f;
    for (int i = threadIdx.x; i < n; i += 256) s += partials[i];  // fixed order
    sdata[threadIdx.x] = s;
    __syncthreads();
    for (int off = 128; off > 0; off >>= 1) {
        if (threadIdx.x < (unsigned)off) sdata[threadIdx.x] += sdata[threadIdx.x + off];
        __syncthreads();
    }
    if (threadIdx.x == 0) out[0] = sdata[0];
}

// ---------------------------------------------------------------------------
extern "C" void kernel_launch(void* const* d_in, const int* in_sizes, int n_in,
                              void* d_out, int out_size, void* d_ws, size_t ws_size,
                              hipStream_t stream) {
    const float* x     = (const float*)d_in[0];       // (32,8,256,256) f32
    const float* nodal = (const float*)d_in[1];       // (8,128) f32
    float* out = (float*)d_out;                       // scalar

    const int nx = in_sizes[0];
    const int n4 = nx / 4;

    // workspace layout: [0, 16KB) coefficient table, then per-block partials
    float4* coef     = (float4*)d_ws;
    float*  partials = (float*)((char*)d_ws + (size_t)(NM * CO_STRIDE) * sizeof(float4));

    spline_coeff_kernel<<<1, 32, 0, stream>>>(nodal, coef);
    spline_eval_kernel<<<EVAL_BLOCKS, EVAL_THREADS, 0, stream>>>(x, coef, partials, n4);
    spline_reduce_kernel<<<1, 256, 0, stream>>>(partials, EVAL_BLOCKS, out);
}